// GAT_34600256536667
// MI455X (gfx1250) — compile-verified
//
#include <hip/hip_runtime.h>
#include <hip/hip_bf16.h>

// Sizes fixed by the reference problem
#define NG   5
#define CB   100   // chunks (logical)
#define PC   112   // chunks padded to 7 full 16-row WMMA tiles
#define BB   8
#define SS   500
#define UU   128
#define FF   20    // 2 * C(5,2)
#define HH   5
#define GHH  4
#define GDD  32

typedef float v2f __attribute__((ext_vector_type(2)));
typedef float v8f __attribute__((ext_vector_type(8)));

__device__ __forceinline__ v8f wmma4(v2f a, v2f b, v8f c) {
  // D(16x16,f32) = A(16x4,f32) x B(4x16,f32) + C
  return __builtin_amdgcn_wmma_f32_16x16x4_f32(false, a, false, b, (short)0, c,
                                               false, false);
}

// ---------------------------------------------------------------------------
// Kernel 1: per-chunk edge MHA -> g[b,c,u]  (VALU; dk=2, F=20 too small for WMMA)
// grid = B*PC blocks, 128 threads (one per u); pad chunks write zeros.
// ---------------------------------------------------------------------------
__global__ __launch_bounds__(128) void k_chunk_mha(
    const float* __restrict__ x,
    const float* __restrict__ wq, const float* __restrict__ bq,
    const float* __restrict__ wk, const float* __restrict__ bk,
    const float* __restrict__ wv, const float* __restrict__ bv,
    const float* __restrict__ wo, const float* __restrict__ bo,
    const float* __restrict__ pos, float* __restrict__ g_out)
{
  const int bc = blockIdx.x;
  const int b = bc / PC, c = bc % PC;
  const int u = threadIdx.x;

  if (c >= CB) {               // block-uniform: zero-fill pad rows of g
    g_out[((size_t)b * PC + c) * UU + u] = 0.f;
    return;
  }

  __shared__ float nodes[NG][UU];
  __shared__ float sk[UU][10];
  __shared__ float sv[UU][10];
  __shared__ float swq[FF * 10], swk[FF * 10], swv[FF * 10];
  __shared__ float swo[10 * FF];
  __shared__ float sbq[10], sbk[10], sbv[10], sbo[FF];

#pragma unroll
  for (int t = 0; t < NG; ++t)
    nodes[t][u] = x[((size_t)b * SS + (size_t)c * NG + t) * UU + u];
  for (int i = u; i < FF * 10; i += 128) {
    swq[i] = wq[c * FF * 10 + i];
    swk[i] = wk[c * FF * 10 + i];
    swv[i] = wv[c * FF * 10 + i];
    swo[i] = wo[c * 10 * FF + i];
  }
  if (u < 10) { sbq[u] = bq[c * 10 + u]; sbk[u] = bk[c * 10 + u]; sbv[u] = bv[c * 10 + u]; }
  if (u < FF) sbo[u] = bo[c * FF + u];
  __syncthreads();

  // edge concat pattern: f=2p -> ii[p], f=2p+1 -> jj[p]
  const int nodeOf[FF] = {0,1, 0,2, 0,3, 0,4, 1,2, 1,3, 1,4, 2,3, 2,4, 3,4};
  float ht[FF];
#pragma unroll
  for (int f = 0; f < FF; ++f) ht[f] = nodes[nodeOf[f]][u];

  float q[10];
  const float rs2 = 0.70710678118654752f; // 1/sqrt(dk=2)
#pragma unroll
  for (int hd = 0; hd < 10; ++hd) {
    float aq = sbq[hd], ak = sbk[hd], av = sbv[hd];
#pragma unroll
    for (int f = 0; f < FF; ++f) {
      float h = ht[f];
      aq += h * swq[f * 10 + hd];
      ak += h * swk[f * 10 + hd];
      av += h * swv[f * 10 + hd];
    }
    q[hd] = aq * rs2;
    sk[u][hd] = ak;
    sv[u][hd] = av;
  }
  __syncthreads();

  float o[10];
#pragma unroll
  for (int h = 0; h < HH; ++h) {
    const float q0 = q[h * 2], q1 = q[h * 2 + 1];
    float m = -3.4e38f;
    for (int vv = 0; vv < UU; ++vv) {
      float s = q0 * sk[vv][h * 2] + q1 * sk[vv][h * 2 + 1];
      m = fmaxf(m, s);
    }
    float sum = 0.f, o0 = 0.f, o1 = 0.f;
    for (int vv = 0; vv < UU; ++vv) {
      float s = q0 * sk[vv][h * 2] + q1 * sk[vv][h * 2 + 1];
      float p = __expf(s - m);
      sum += p;
      o0 += p * sv[vv][h * 2];
      o1 += p * sv[vv][h * 2 + 1];
    }
    float inv = 1.0f / sum;
    o[h * 2] = o0 * inv; o[h * 2 + 1] = o1 * inv;
  }

  float acc = 0.f;
#pragma unroll
  for (int f = 0; f < FF; ++f) {
    float mo = sbo[f];
#pragma unroll
    for (int hd = 0; hd < 10; ++hd) mo += o[hd] * swo[hd * FF + f];
    float a = ht[f] + mo;
    acc += (a >= 0.f) ? a : 0.3f * a; // leaky_relu(0.3)
  }
  g_out[((size_t)b * PC + c) * UU + u] = acc * (1.0f / 20.0f) + pos[c * UU + u];
}

// ---------------------------------------------------------------------------
// Kernel 2: gq/gk/gv = g[b] (112x128) @ W (128x128) + bias   (WMMA f32 16x16x4)
// grid = B*7 row tiles, 128 threads = 4 waves x 2 col tiles; unconditional I/O
// ---------------------------------------------------------------------------
__global__ __launch_bounds__(128) void k_gqkv(
    const float* __restrict__ g,
    const float* __restrict__ gwq, const float* __restrict__ gbq,
    const float* __restrict__ gwk, const float* __restrict__ gbk,
    const float* __restrict__ gwv, const float* __restrict__ gbv,
    float* __restrict__ gq, float* __restrict__ gk, float* __restrict__ gv)
{
  const int b = blockIdx.x / 7, mt = blockIdx.x % 7;
  const int row0 = mt * 16;
  __shared__ float sA[16 * UU];
  {
    const float4* src = (const float4*)(g + ((size_t)b * PC + row0) * UU);
    float4* dst = (float4*)sA;
    for (int i = threadIdx.x; i < 16 * UU / 4; i += 128) dst[i] = src[i];
  }
  __syncthreads();

  const int lane = threadIdx.x & 31, wid = threadIdx.x >> 5;
  const int m = lane & 15;
  const int kb2 = (lane >> 4) << 1;   // K sub-offset per half-wave
  const int rsel = (lane >> 4) << 3;  // D row offset per half-wave
  const float gscale = 0.17677669529663688f; // 1/sqrt(32)

  for (int rep = 0; rep < 2; ++rep) {
    const int n0 = (wid * 2 + rep) * 16;
    v8f aq = {}, ak = {}, av = {};
    for (int ks = 0; ks < 32; ++ks) {
      const int kb = ks * 4 + kb2;
      v2f A; A.x = sA[m * UU + kb]; A.y = sA[m * UU + kb + 1];
      v2f Bq, Bk, Bv;
      Bq.x = gwq[kb * UU + n0 + m];       Bq.y = gwq[(kb + 1) * UU + n0 + m];
      Bk.x = gwk[kb * UU + n0 + m];       Bk.y = gwk[(kb + 1) * UU + n0 + m];
      Bv.x = gwv[kb * UU + n0 + m];       Bv.y = gwv[(kb + 1) * UU + n0 + m];
      aq = wmma4(A, Bq, aq);
      ak = wmma4(A, Bk, ak);
      av = wmma4(A, Bv, av);
    }
    const int col = n0 + m;
    const float biasq = gbq[col], biask = gbk[col], biasv = gbv[col];
#pragma unroll
    for (int r = 0; r < 8; ++r) {
      const int row = row0 + r + rsel;
      const size_t off = ((size_t)b * PC + row) * UU + col;
      gq[off] = (aq[r] + biasq) * gscale;
      gk[off] = ak[r] + biask;
      gv[off] = av[r] + biasv;
    }
  }
}

// ---------------------------------------------------------------------------
// Kernel 3: gs[b,h] = gq (112x32) @ gk^T (32x112)   (WMMA; K=32 -> 8 steps)
// grid = B*GH*7 (q tiles), 224 threads = 7 waves, one kc tile per wave
// ---------------------------------------------------------------------------
__global__ __launch_bounds__(224) void k_gscores(
    const float* __restrict__ gq, const float* __restrict__ gk, float* __restrict__ gs)
{
  const int qt = blockIdx.x % 7;
  const int bh = blockIdx.x / 7;
  const int b = bh / GHH, h = bh % GHH;
  const int lane = threadIdx.x & 31;
  const int nt = threadIdx.x >> 5; // kc tile 0..6
  const int m = lane & 15;
  const int kb2 = (lane >> 4) << 1;
  const int rsel = (lane >> 4) << 3;

  const int rowA = qt * 16 + m;   // padded rows always valid to read
  const int colB = nt * 16 + m;
  const float* gqp = gq + ((size_t)b * PC + rowA) * UU + h * GDD;
  const float* gkp = gk + ((size_t)b * PC + colB) * UU + h * GDD;

  v8f acc = {};
#pragma unroll
  for (int ks = 0; ks < 8; ++ks) {
    const int kb = ks * 4 + kb2;
    v2f A; A.x = gqp[kb]; A.y = gqp[kb + 1];
    v2f B; B.x = gkp[kb]; B.y = gkp[kb + 1];
    acc = wmma4(A, B, acc);
  }
  const int kc = nt * 16 + m;
#pragma unroll
  for (int r = 0; r < 8; ++r) {
    const int row = qt * 16 + r + rsel;
    gs[(((size_t)b * GHH + h) * PC + row) * PC + kc] = acc[r];
  }
}

// ---------------------------------------------------------------------------
// Kernel 4: row softmax over kc (first 100 of 112); one block per logical row
// ---------------------------------------------------------------------------
__global__ __launch_bounds__(128) void k_softmax(float* __restrict__ gs)
{
  __shared__ float red[128];
  const int bh = blockIdx.x / CB;
  const int qc = blockIdx.x % CB;
  float* p = gs + ((size_t)bh * PC + qc) * PC;
  const int t = threadIdx.x;
  float v = (t < CB) ? p[t] : -3.4e38f;
  red[t] = v;
  __syncthreads();
  for (int s = 64; s > 0; s >>= 1) {
    if (t < s) red[t] = fmaxf(red[t], red[t + s]);
    __syncthreads();
  }
  const float m = red[0];
  __syncthreads();
  float e = (t < CB) ? __expf(v - m) : 0.f;
  red[t] = e;
  __syncthreads();
  for (int s = 64; s > 0; s >>= 1) {
    if (t < s) red[t] += red[t + s];
    __syncthreads();
  }
  const float inv = 1.0f / red[0];
  if (t < CB) p[t] = e * inv;
}

// ---------------------------------------------------------------------------
// Kernel 5: go[b,h] = ga (112x100) @ gv (100x32)   (WMMA; K=100 -> 25 steps)
// grid = B*GH, 448 threads = 14 waves = 7 q-tiles x 2 n-tiles
// ---------------------------------------------------------------------------
__global__ __launch_bounds__(448) void k_go(
    const float* __restrict__ ga, const float* __restrict__ gv, float* __restrict__ go)
{
  const int b = blockIdx.x / GHH, h = blockIdx.x % GHH;
  const int lane = threadIdx.x & 31;
  const int wid = threadIdx.x >> 5;
  const int qt = wid >> 1, nt = wid & 1;
  const int m = lane & 15;
  const int kb2 = (lane >> 4) << 1;
  const int rsel = (lane >> 4) << 3;

  const int rowA = qt * 16 + m;
  const float* gap = ga + (((size_t)b * GHH + h) * PC + rowA) * PC;
  const float* gvp = gv + ((size_t)b * PC) * UU + h * GDD + nt * 16 + m;

  v8f acc = {};
  for (int ks = 0; ks < 25; ++ks) {   // K = 100 exactly; softmax cols only
    const int kb = ks * 4 + kb2;
    v2f A; A.x = gap[kb]; A.y = gap[kb + 1];
    v2f B; B.x = gvp[(size_t)kb * UU]; B.y = gvp[(size_t)(kb + 1) * UU];
    acc = wmma4(A, B, acc);
  }
#pragma unroll
  for (int r = 0; r < 8; ++r) {
    const int row = qt * 16 + r + rsel;
    go[((size_t)b * PC + row) * UU + h * GDD + nt * 16 + m] = acc[r];
  }
}

// ---------------------------------------------------------------------------
// Kernel 6: out = g + go (112x128) @ gwo (128x128) + gbo   (WMMA; fused residual)
// grid = B*7, 128 threads = 4 waves x 2 col tiles; only d_out store is masked
// ---------------------------------------------------------------------------
__global__ __launch_bounds__(128) void k_out(
    const float* __restrict__ go, const float* __restrict__ gwo,
    const float* __restrict__ gbo, const float* __restrict__ g,
    float* __restrict__ out)
{
  const int b = blockIdx.x / 7, mt = blockIdx.x % 7;
  const int row0 = mt * 16;
  __shared__ float sA[16 * UU];
  {
    const float4* src = (const float4*)(go + ((size_t)b * PC + row0) * UU);
    float4* dst = (float4*)sA;
    for (int i = threadIdx.x; i < 16 * UU / 4; i += 128) dst[i] = src[i];
  }
  __syncthreads();

  const int lane = threadIdx.x & 31, wid = threadIdx.x >> 5;
  const int m = lane & 15;
  const int kb2 = (lane >> 4) << 1;
  const int rsel = (lane >> 4) << 3;
  const bool fullTile = (row0 + 16 <= CB);   // block-uniform

  for (int rep = 0; rep < 2; ++rep) {
    const int n0 = (wid * 2 + rep) * 16;
    v8f acc = {};
    for (int ks = 0; ks < 32; ++ks) {
      const int kb = ks * 4 + kb2;
      v2f A; A.x = sA[m * UU + kb]; A.y = sA[m * UU + kb + 1];
      v2f B; B.x = gwo[kb * UU + n0 + m]; B.y = gwo[(kb + 1) * UU + n0 + m];
      acc = wmma4(A, B, acc);
    }
    const int col = n0 + m;
    const float bias = gbo[col];
    if (fullTile) {
#pragma unroll
      for (int r = 0; r < 8; ++r) {
        const int row = row0 + r + rsel;
        out[((size_t)b * CB + row) * UU + col] =
            g[((size_t)b * PC + row) * UU + col] + acc[r] + bias;
      }
    } else {
#pragma unroll
      for (int r = 0; r < 8; ++r) {
        const int row = row0 + r + rsel;
        if (row < CB)
          out[((size_t)b * CB + row) * UU + col] =
              g[((size_t)b * PC + row) * UU + col] + acc[r] + bias;
      }
    }
  }
}

// ---------------------------------------------------------------------------
extern "C" void kernel_launch(void* const* d_in, const int* in_sizes, int n_in,
                              void* d_out, int out_size, void* d_ws, size_t ws_size,
                              hipStream_t stream)
{
  (void)in_sizes; (void)n_in; (void)out_size; (void)ws_size;
  const float* x   = (const float*)d_in[0];
  const float* wq  = (const float*)d_in[1];
  const float* bq  = (const float*)d_in[2];
  const float* wk  = (const float*)d_in[3];
  const float* bk  = (const float*)d_in[4];
  const float* wv  = (const float*)d_in[5];
  const float* bv  = (const float*)d_in[6];
  const float* wo  = (const float*)d_in[7];
  const float* bo  = (const float*)d_in[8];
  const float* pos = (const float*)d_in[9];
  const float* gwq = (const float*)d_in[10];
  const float* gbq = (const float*)d_in[11];
  const float* gwk = (const float*)d_in[12];
  const float* gbk = (const float*)d_in[13];
  const float* gwv = (const float*)d_in[14];
  const float* gbv = (const float*)d_in[15];
  const float* gwo = (const float*)d_in[16];
  const float* gbo = (const float*)d_in[17];
  float* out = (float*)d_out;
  float* ws  = (float*)d_ws;

  const size_t NBU = (size_t)BB * PC * UU;          // 114688 padded [B,112,128]
  float* g_buf  = ws;
  float* gq_buf = ws + NBU;
  float* gk_buf = ws + 2 * NBU;
  float* gv_buf = ws + 3 * NBU;
  float* gs_buf = ws + 4 * NBU;                     // [B,GH,112,112] = 401408
  float* go_buf = gs_buf + (size_t)BB * GHH * PC * PC;

  k_chunk_mha<<<BB * PC, 128, 0, stream>>>(x, wq, bq, wk, bk, wv, bv, wo, bo,
                                           pos, g_buf);
  k_gqkv<<<BB * 7, 128, 0, stream>>>(g_buf, gwq, gbq, gwk, gbk, gwv, gbv,
                                     gq_buf, gk_buf, gv_buf);
  k_gscores<<<BB * GHH * 7, 224, 0, stream>>>(gq_buf, gk_buf, gs_buf);
  k_softmax<<<BB * GHH * CB, 128, 0, stream>>>(gs_buf);
  k_go<<<BB * GHH, 448, 0, stream>>>(gs_buf, gv_buf, go_buf);
  k_out<<<BB * 7, 128, 0, stream>>>(go_buf, gwo, gbo, g_buf, out);
}